// LocalizationLoss_47553877901940
// MI455X (gfx1250) — compile-verified
//
#include <hip/hip_runtime.h>

// ---------------------------------------------------------------------------
// LocalizationLoss (Hungarian matching over 3 objects) for gfx1250 / MI455X.
// Stage 1: bandwidth-bound streaming kernel, b128 loads -> LDS staging,
//          per-thread cost matrix + 6-perm argmin + loss, wave32 reduction.
// Stage 2: single-wave reduction of 2048 block partials on the matrix pipe
//          using V_WMMA_F32_16X16X4_F32 with a ones A-matrix (column-sum
//          trick; layout-independent total).
// ---------------------------------------------------------------------------

typedef __attribute__((ext_vector_type(2))) float v2f;
typedef __attribute__((ext_vector_type(8))) float v8f;

#define NB    524288      // B from reference
#define TPB   256         // samples (=threads) per block
#define NBLK  (NB / TPB)  // 2048 partials

__device__ __forceinline__ float bce_f(float p, float t) {
    // -(t*clip(log p, -100) + (1-t)*clip(log1p(-p), -100))
    float lp = fmaxf(logf(p), -100.0f);
    float lq = fmaxf(log1pf(-p), -100.0f);
    return -(t * lp + (1.0f - t) * lq);
}

__global__ __launch_bounds__(TPB)
void loc_loss_stage1(const float* __restrict__ outp,
                     const float* __restrict__ tgt,
                     float* __restrict__ partials) {
    // Block chunk: 256 samples * 21 floats = 21504 B (16B multiple),
    //              256 samples * 15 floats = 15360 B (16B multiple).
    __shared__ float4 s_o4[TPB * 21 / 4];   // 1344 float4
    __shared__ float4 s_t4[TPB * 15 / 4];   //  960 float4
    __shared__ float  s_w[TPB / 32];

    const int tid = threadIdx.x;
    const int blk = blockIdx.x;

    const float4* go = (const float4*)(outp + (size_t)blk * TPB * 21);
    const float4* gt = (const float4*)(tgt  + (size_t)blk * TPB * 15);
    #pragma unroll 2
    for (int i = tid; i < TPB * 21 / 4; i += TPB) s_o4[i] = go[i];
    #pragma unroll 2
    for (int i = tid; i < TPB * 15 / 4; i += TPB) s_t4[i] = gt[i];
    __syncthreads();

    // Per-sample reads: strides 21 and 15 are coprime with 64 banks.
    const float* so = (const float*)s_o4 + tid * 21;
    const float* st = (const float*)s_t4 + tid * 15;

    float p[3], pb[3][3], lg[3][3];
    float c[3], tb[3][3];
    int   k[3];
    #pragma unroll
    for (int i = 0; i < 3; ++i) {
        p[i]     = so[i * 7 + 0];
        pb[i][0] = so[i * 7 + 1]; pb[i][1] = so[i * 7 + 2]; pb[i][2] = so[i * 7 + 3];
        lg[i][0] = so[i * 7 + 4]; lg[i][1] = so[i * 7 + 5]; lg[i][2] = so[i * 7 + 6];
        c[i]     = st[i * 5 + 0];
        tb[i][0] = st[i * 5 + 1]; tb[i][1] = st[i * 5 + 2]; tb[i][2] = st[i * 5 + 3];
        k[i]     = (int)st[i * 5 + 4];
    }

    // 3x3 cost matrix: box L2 + class-onehot L2 + |conf diff|
    float tot[3][3];
    #pragma unroll
    for (int i = 0; i < 3; ++i) {
        #pragma unroll
        for (int j = 0; j < 3; ++j) {
            float dx = pb[i][0] - tb[j][0];
            float dy = pb[i][1] - tb[j][1];
            float dz = pb[i][2] - tb[j][2];
            float bd = sqrtf(dx * dx + dy * dy + dz * dz);
            float e0 = lg[i][0] - ((k[j] == 0) ? 1.0f : 0.0f);
            float e1 = lg[i][1] - ((k[j] == 1) ? 1.0f : 0.0f);
            float e2 = lg[i][2] - ((k[j] == 2) ? 1.0f : 0.0f);
            float cd = sqrtf(e0 * e0 + e1 * e1 + e2 * e2);
            float pd = fabsf(p[i] - c[j]);
            tot[i][j] = bd + cd + pd;
        }
    }

    // argmin over the 6 permutations, itertools order, first-min wins (strict <)
    float bc = tot[0][0] + tot[1][1] + tot[2][2]; int m0 = 0, m1 = 1, m2 = 2;
    float cc;
    cc = tot[0][0] + tot[1][2] + tot[2][1]; if (cc < bc) { bc = cc; m0 = 0; m1 = 2; m2 = 1; }
    cc = tot[0][1] + tot[1][0] + tot[2][2]; if (cc < bc) { bc = cc; m0 = 1; m1 = 0; m2 = 2; }
    cc = tot[0][1] + tot[1][2] + tot[2][0]; if (cc < bc) { bc = cc; m0 = 1; m1 = 2; m2 = 0; }
    cc = tot[0][2] + tot[1][0] + tot[2][1]; if (cc < bc) { bc = cc; m0 = 2; m1 = 0; m2 = 1; }
    cc = tot[0][2] + tot[1][1] + tot[2][0]; if (cc < bc) { bc = cc; m0 = 2; m1 = 1; m2 = 0; }

    // unmatched fallback: mean elementwise BCE
    float bfull = (bce_f(p[0], c[0]) + bce_f(p[1], c[1]) + bce_f(p[2], c[2])) * (1.0f / 3.0f);

    // log-sum-exp per prediction row
    float lse[3];
    #pragma unroll
    for (int i = 0; i < 3; ++i) {
        float mx = fmaxf(lg[i][0], fmaxf(lg[i][1], lg[i][2]));
        lse[i] = mx + logf(expf(lg[i][0] - mx) + expf(lg[i][1] - mx) + expf(lg[i][2] - mx));
    }

    float loss = 0.0f;
    #pragma unroll
    for (int i = 0; i < 3; ++i) {
        int   j   = (i == 0) ? m0 : ((i == 1) ? m1 : m2);
        float tc  = (j == 0) ? c[0] : ((j == 1) ? c[1] : c[2]);
        float b0  = (j == 0) ? tb[0][0] : ((j == 1) ? tb[1][0] : tb[2][0]);
        float b1  = (j == 0) ? tb[0][1] : ((j == 1) ? tb[1][1] : tb[2][1]);
        float b2  = (j == 0) ? tb[0][2] : ((j == 1) ? tb[1][2] : tb[2][2]);
        int   kk  = (j == 0) ? k[0] : ((j == 1) ? k[1] : k[2]);
        float d0 = pb[i][0] - b0, d1 = pb[i][1] - b1, d2 = pb[i][2] - b2;
        float mse  = (d0 * d0 + d1 * d1 + d2 * d2) * (1.0f / 3.0f);
        float bp   = bce_f(p[i], tc);
        float lsel = (kk == 0) ? lg[i][0] : ((kk == 1) ? lg[i][1] : lg[i][2]);
        float ce   = lse[i] - lsel;
        float matched = mse + bp + ce;
        loss += (tc > 0.0f) ? matched : bfull;   // branchless select, no divergence
    }

    // wave32 reduction (warpSize == 32 on gfx1250)
    #pragma unroll
    for (int m = 16; m > 0; m >>= 1) loss += __shfl_xor(loss, m, 32);
    if ((tid & 31) == 0) s_w[tid >> 5] = loss;
    __syncthreads();
    if (tid == 0) {
        float s = 0.0f;
        #pragma unroll
        for (int w = 0; w < TPB / 32; ++w) s += s_w[w];
        partials[blk] = s;
    }
}

// Stage 2: one wave, EXEC all-ones (WMMA requirement). A = ones(16x4), so
// D[m,n] = sum_k B[k,n]; accumulating over 32 chunks of 64 partials sums
// everything regardless of the logical lane<->element mapping. Rows 0 and 8
// (lanes 0-15 / 16-31 of acc[0]) each carry the full set of column sums, so a
// 32-lane shfl_xor collapse yields 2x the grand total.
__global__ __launch_bounds__(32)
void loc_loss_stage2(const float* __restrict__ partials, float* __restrict__ outp) {
    const int lane = threadIdx.x;
    v2f a; a[0] = 1.0f; a[1] = 1.0f;
    v8f acc = {};
    for (int it = 0; it < NBLK / 64; ++it) {
        v2f b;
        b[0] = partials[it * 64 + lane * 2 + 0];
        b[1] = partials[it * 64 + lane * 2 + 1];
        acc = __builtin_amdgcn_wmma_f32_16x16x4_f32(
            /*neg_a=*/false, a, /*neg_b=*/false, b,
            /*c_mod=*/(short)0, acc, /*reuse_a=*/false, /*reuse_b=*/false);
    }
    float r = acc[0];
    #pragma unroll
    for (int m = 16; m > 0; m >>= 1) r += __shfl_xor(r, m, 32);
    if (lane == 0) outp[0] = r * (0.5f / (float)NB);
}

extern "C" void kernel_launch(void* const* d_in, const int* in_sizes, int n_in,
                              void* d_out, int out_size, void* d_ws, size_t ws_size,
                              hipStream_t stream) {
    const float* outp = (const float*)d_in[0];   // (B, 3, 7) f32
    const float* tgt  = (const float*)d_in[1];   // (B, 3, 5) f32
    float* partials   = (float*)d_ws;            // NBLK floats = 8 KB
    loc_loss_stage1<<<NBLK, TPB, 0, stream>>>(outp, tgt, partials);
    loc_loss_stage2<<<1, 32, 0, stream>>>(partials, (float*)d_out);
}